// ColorJitterRandomizerPerSample_20779051778684
// MI455X (gfx1250) — compile-verified
//
#include <hip/hip_runtime.h>
#include <cstdint>

// ---------------------------------------------------------------------------
// CDNA5 (gfx1250) async global->LDS path, guarded for compile portability.
// ---------------------------------------------------------------------------
#define AS1 __attribute__((address_space(1)))
#define AS3 __attribute__((address_space(3)))

typedef int v4i __attribute__((vector_size(16)));      // builtin's pointee type
typedef float v4f __attribute__((ext_vector_type(4))); // native vector for NT stores

#if defined(__gfx1250__) && __has_builtin(__builtin_amdgcn_global_load_async_to_lds_b128)
#define HAVE_ASYNC_LDS 1
#else
#define HAVE_ASYNC_LDS 0
#endif

#if HAVE_ASYNC_LDS
#if __has_builtin(__builtin_amdgcn_s_wait_asynccnt)
#define WAIT_ASYNC(n) __builtin_amdgcn_s_wait_asynccnt(n)
#else
#define WAIT_ASYNC(n) asm volatile("s_wait_asynccnt %0" ::"i"(n) : "memory")
#endif
#else
#define WAIT_ASYNC(n) ((void)0)
#endif

__device__ __forceinline__ void async_copy_b128(const float* __restrict__ g, float* l) {
#if HAVE_ASYNC_LDS
  // 16 bytes per lane, global -> LDS, bypassing VGPRs (tracked by ASYNCcnt).
  __builtin_amdgcn_global_load_async_to_lds_b128((AS1 v4i*)g, (AS3 v4i*)l, 0, 0);
#else
  *(float4*)l = *(const float4*)g;
#endif
}

// Non-temporal 128-bit store: keep the 201MB output stream from evicting the
// L2-resident input that pass 2 still needs (lowered with th:TH_STORE_NT).
__device__ __forceinline__ void nt_store_b128(float* p, float x, float y, float z, float w) {
#if __has_builtin(__builtin_nontemporal_store)
  v4f v = {x, y, z, w};
  __builtin_nontemporal_store(v, (v4f*)p);
#else
  float4 v = make_float4(x, y, z, w);
  *(float4*)p = v;
#endif
}

__device__ __forceinline__ float clip01(float x) { return fminf(fmaxf(x, 0.0f), 1.0f); }

// ---------------------------------------------------------------------------
// Kernel A: per-image mean of grayscale(clip(img * brightness)).
// One block per image, async double-buffered streaming through LDS.
// ---------------------------------------------------------------------------
#define TPB_A 512

__global__ __launch_bounds__(TPB_A) void cj_mean_kernel(
    const float* __restrict__ in, const float* __restrict__ brightness,
    float* __restrict__ mean_out, int hw) {
  const int b = blockIdx.x;
  const int t = threadIdx.x;
  const float bf = brightness[b];
  const float* base = in + (size_t)b * 3u * (size_t)hw;

  __shared__ __align__(16) float lbuf[2][3][TPB_A * 4];  // 48 KB (2 buf x 3 ch x 8 KB)
  __shared__ float red[TPB_A];

  const int tile_elems = TPB_A * 4;
  const int iters = hw / tile_elems;  // 65536 / 2048 = 32 for this workload

  float accr = 0.f, accg = 0.f, accb = 0.f;

  // Prologue: tile 0 -> buffer 0
  {
    const float* p = base + t * 4;
    async_copy_b128(p,           &lbuf[0][0][t * 4]);
    async_copy_b128(p + hw,      &lbuf[0][1][t * 4]);
    async_copy_b128(p + 2 * hw,  &lbuf[0][2][t * 4]);
  }

  for (int it = 0; it < iters; ++it) {
    const int buf = it & 1;
    if (it + 1 < iters) {
      // Prefetch next tile into the other buffer, then wait for current tile
      // (async loads complete in order: <=3 outstanding means ours are done).
      const float* p = base + (size_t)(it + 1) * tile_elems + t * 4;
      async_copy_b128(p,           &lbuf[buf ^ 1][0][t * 4]);
      async_copy_b128(p + hw,      &lbuf[buf ^ 1][1][t * 4]);
      async_copy_b128(p + 2 * hw,  &lbuf[buf ^ 1][2][t * 4]);
      WAIT_ASYNC(3);
    } else {
      WAIT_ASYNC(0);
    }
    __syncthreads();  // current tile visible to all waves

    float4 rv = *(const float4*)&lbuf[buf][0][t * 4];
    float4 gv = *(const float4*)&lbuf[buf][1][t * 4];
    float4 bv = *(const float4*)&lbuf[buf][2][t * 4];
    accr += clip01(rv.x * bf) + clip01(rv.y * bf) + clip01(rv.z * bf) + clip01(rv.w * bf);
    accg += clip01(gv.x * bf) + clip01(gv.y * bf) + clip01(gv.z * bf) + clip01(gv.w * bf);
    accb += clip01(bv.x * bf) + clip01(bv.y * bf) + clip01(bv.z * bf) + clip01(bv.w * bf);

    __syncthreads();  // all reads of this buffer done before it is refilled
  }

  // clip is per-element, so gray weighting is separable per channel.
  red[t] = 0.2989f * accr + 0.587f * accg + 0.114f * accb;
  __syncthreads();
  for (int s = TPB_A / 2; s > 0; s >>= 1) {
    if (t < s) red[t] += red[t + s];
    __syncthreads();
  }
  if (t == 0) mean_out[b] = red[0] / (float)hw;
}

// ---------------------------------------------------------------------------
// Per-pixel jitter math (branchless; ternaries lower to v_cndmask).
// ---------------------------------------------------------------------------
__device__ __forceinline__ void jitter_pixel(float r, float g, float bl,
                                             float bf, float cf, float sf,
                                             float hf, float m,
                                             float& ro, float& go, float& bo) {
  // brightness
  r = clip01(r * bf); g = clip01(g * bf); bl = clip01(bl * bf);
  // contrast blend toward per-image gray mean
  r  = clip01(cf * r  + (1.f - cf) * m);
  g  = clip01(cf * g  + (1.f - cf) * m);
  bl = clip01(cf * bl + (1.f - cf) * m);
  // saturation blend toward this pixel's gray
  const float gray = 0.2989f * r + 0.587f * g + 0.114f * bl;
  r  = clip01(sf * r  + (1.f - sf) * gray);
  g  = clip01(sf * g  + (1.f - sf) * gray);
  bl = clip01(sf * bl + (1.f - sf) * gray);

  // rgb -> hsv
  const float maxc = fmaxf(r, fmaxf(g, bl));
  const float minc = fminf(r, fminf(g, bl));
  const bool eqc = (maxc == minc);
  const float cr = maxc - minc;
  const float s = cr / (eqc ? 1.f : maxc);
  const float crd = eqc ? 1.f : cr;
  const float rc = (maxc - r) / crd;
  const float gc = (maxc - g) / crd;
  const float bc = (maxc - bl) / crd;
  const float hr = (maxc == r) ? (bc - gc) : 0.f;
  const float hg = ((maxc == g) && (maxc != r)) ? (2.f + rc - bc) : 0.f;
  const float hb = ((maxc != g) && (maxc != r)) ? (4.f + gc - rc) : 0.f;
  float h = (hr + hg + hb) * (1.f / 6.f) + 1.f;
  h = h - floorf(h);          // jnp.mod(x, 1)
  h = h + hf;                 // hue shift
  h = h - floorf(h);

  // hsv -> rgb
  const float v = maxc;
  const float h6 = h * 6.f;
  const float fi = floorf(h6);
  const float f = h6 - fi;
  int i = ((int)fi) % 6;      // fi in [0,6]; %6 maps 6 -> 0 like the reference
  const float pp = clip01(v * (1.f - s));
  const float q  = clip01(v * (1.f - s * f));
  const float tt = clip01(v * (1.f - s * (1.f - f)));
  // a1=[v,q,p,p,t,v], a2=[t,v,v,q,p,p], a3=[p,p,t,v,v,q]
  ro = ((i == 0) | (i == 5)) ? v : (i == 1) ? q : (i == 4) ? tt : pp;
  go = ((i == 1) | (i == 2)) ? v : (i == 0) ? tt : (i == 3) ? q : pp;
  bo = ((i == 3) | (i == 4)) ? v : (i == 2) ? tt : (i == 5) ? q : pp;
}

// ---------------------------------------------------------------------------
// Kernel B: elementwise transform, float4 (B128) planar loads, NT B128 stores.
// Grid: (hw / (TPB_B*4), B). Input re-read should largely hit the 192MB L2;
// NT stores keep the output stream from evicting it.
// ---------------------------------------------------------------------------
#define TPB_B 256

__global__ __launch_bounds__(TPB_B) void cj_apply_kernel(
    const float* __restrict__ in, const float* __restrict__ brightness,
    const float* __restrict__ contrast, const float* __restrict__ saturation,
    const float* __restrict__ hue, const float* __restrict__ mean_in,
    float* __restrict__ out, int hw) {
  const int b = blockIdx.y;
  const int p = (blockIdx.x * TPB_B + (int)threadIdx.x) * 4;
  if (p >= hw) return;
  // Uniform per-image scalars -> SGPRs
  const float bf = brightness[b];
  const float cf = contrast[b];
  const float sf = saturation[b];
  const float hf = hue[b];
  const float m  = mean_in[b];

  const size_t base = (size_t)b * 3u * (size_t)hw + (size_t)p;
  float4 rv = *(const float4*)(in + base);
  float4 gv = *(const float4*)(in + base + hw);
  float4 bv = *(const float4*)(in + base + 2 * hw);

  float4 ro, go, bo;
  jitter_pixel(rv.x, gv.x, bv.x, bf, cf, sf, hf, m, ro.x, go.x, bo.x);
  jitter_pixel(rv.y, gv.y, bv.y, bf, cf, sf, hf, m, ro.y, go.y, bo.y);
  jitter_pixel(rv.z, gv.z, bv.z, bf, cf, sf, hf, m, ro.z, go.z, bo.z);
  jitter_pixel(rv.w, gv.w, bv.w, bf, cf, sf, hf, m, ro.w, go.w, bo.w);

  nt_store_b128(out + base,          ro.x, ro.y, ro.z, ro.w);
  nt_store_b128(out + base + hw,     go.x, go.y, go.z, go.w);
  nt_store_b128(out + base + 2 * hw, bo.x, bo.y, bo.z, bo.w);
}

// ---------------------------------------------------------------------------
// Host launcher
// ---------------------------------------------------------------------------
extern "C" void kernel_launch(void* const* d_in, const int* in_sizes, int n_in,
                              void* d_out, int out_size, void* d_ws, size_t ws_size,
                              hipStream_t stream) {
  const float* in = (const float*)d_in[0];
  const float* bf = (const float*)d_in[1];
  const float* cf = (const float*)d_in[2];
  const float* sf = (const float*)d_in[3];
  const float* hf = (const float*)d_in[4];
  float* out = (float*)d_out;
  float* mean = (float*)d_ws;  // B floats of scratch

  const int B = in_sizes[1];                 // brightness_f has B elements
  const int hw = in_sizes[0] / (3 * B);      // H*W per image (65536)

  cj_mean_kernel<<<dim3(B), dim3(TPB_A), 0, stream>>>(in, bf, mean, hw);

  dim3 gridB((hw + TPB_B * 4 - 1) / (TPB_B * 4), B);
  cj_apply_kernel<<<gridB, dim3(TPB_B), 0, stream>>>(in, bf, cf, sf, hf, mean, out, hw);
}